// StructuredRotation_44392781971752
// MI455X (gfx1250) — compile-verified
//
#include <hip/hip_runtime.h>

// CDNA5 / gfx1250: wave32, V_WMMA_F32_16X16X4_F32 (f32 matrix cores).
typedef __attribute__((ext_vector_type(2))) float v2f;
typedef __attribute__((ext_vector_type(8))) float v8f;

#define DH       64
#define STR      65            // LDS row stride in floats (pad -> conflict-free column walks)
#define NTERM    8             // Taylor degree
#define NSQUARE  6             // squarings: exp(A) = exp(A/64)^64
#define SCALE_INV (1.0f/64.0f)

// One 16x16 output tile of D = X(64x64) * Y(64x64), accumulated over K=64
// with 16 chained v_wmma_f32_16x16x4_f32. Xs/Ys are LDS, row stride STR.
// A-frag (16x4, f32): lanes 0-15 -> {K,K+1}, lanes 16-31 -> {K+2,K+3}, M = lane&15.
// B-frag (4x16, f32): lanes 0-15 -> rows {K,K+1}, lanes 16-31 -> rows {K+2,K+3}, N = lane&15.
__device__ __forceinline__ v8f tile_mm64(const float* __restrict__ Xs,
                                         const float* __restrict__ Ys,
                                         int ti, int tj, int lane) {
  v8f c = {0.f, 0.f, 0.f, 0.f, 0.f, 0.f, 0.f, 0.f};
  const int l  = lane & 15;
  const int h2 = (lane >> 4) << 1;  // 0 for lanes 0-15, 2 for lanes 16-31
  const float* xrow = Xs + (ti * 16 + l) * STR + h2;   // walk along K
  const float* ycol = Ys + (tj * 16 + l) + h2 * STR;   // walk down K rows
#pragma unroll
  for (int k = 0; k < DH; k += 4) {
    v2f a, b;
    a.x = xrow[k];
    a.y = xrow[k + 1];
    b.x = ycol[(k + 0) * STR];
    b.y = ycol[(k + 1) * STR];
    // 8 args: (neg_a, A, neg_b, B, c_mod, C, reuse_a, reuse_b)
    c = __builtin_amdgcn_wmma_f32_16x16x4_f32(false, a, false, b, (short)0, c,
                                              false, false);
  }
  return c;
}

// C/D layout: lanes 0-15 hold M=0..7 (vgpr 0..7), lanes 16-31 hold M=8..15. N = lane&15.
__device__ __forceinline__ void tile_store(float* __restrict__ D, int ti, int tj,
                                           int lane, v8f c, float scale) {
  const int l = lane & 15;
  float* drow = D + (ti * 16 + ((lane >> 4) << 3)) * STR + tj * 16 + l;
#pragma unroll
  for (int v = 0; v < 8; ++v) drow[v * STR] = c[v] * scale;
}

__device__ __forceinline__ void tile_acc(float* __restrict__ D, int ti, int tj,
                                         int lane, v8f c, float scale) {
  const int l = lane & 15;
  float* drow = D + (ti * 16 + ((lane >> 4) << 3)) * STR + tj * 16 + l;
#pragma unroll
  for (int v = 0; v < 8; ++v) drow[v * STR] += c[v] * scale;
}

// One workgroup (8 wave32) per (b,s): build X = A/2^6 in LDS, Taylor to deg 8,
// square 6 times, write 64x64 result.
__global__ __launch_bounds__(256)
void structured_rotation_expm_kernel(const float* __restrict__ r,
                                     const float* __restrict__ p0,
                                     const float* __restrict__ p1,
                                     float* __restrict__ out) {
  __shared__ float Xs[DH * STR];  // scaled generator, constant through Taylor
  __shared__ float Ps[DH * STR];  // current Taylor term X^k/k!
  __shared__ float Ss[DH * STR];  // running sum -> result

  const int tid = threadIdx.x;
  const int bs  = blockIdx.x;

  const float r0 = r[bs * 2 + 0];
  const float r1 = r[bs * 2 + 1];

  // Build X = (r0*L0 + r1*L1) * 2^-6 ; P = X ; S = I + X.
  // Skew-symmetric from upper-tri params: idx(i<j) = i*(127-i)/2 + (j-i-1).
#pragma unroll 1
  for (int e = tid; e < DH * DH; e += 256) {
    const int i = e >> 6;
    const int j = e & 63;
    float val = 0.f;
    if (i != j) {
      const int a   = (i < j) ? i : j;
      const int b   = (i < j) ? j : i;
      const int idx = ((a * (2 * DH - 1 - a)) >> 1) + (b - a - 1);
      const float sgn = (i < j) ? 1.f : -1.f;
      val = (r0 * p0[idx] + r1 * p1[idx]) * (sgn * SCALE_INV);
    }
    const int o = i * STR + j;
    Xs[o] = val;
    Ps[o] = val;
    Ss[o] = val + ((i == j) ? 1.f : 0.f);
  }

  const int lane = tid & 31;
  const int wave = tid >> 5;
  const int t0 = wave * 2, t1 = wave * 2 + 1;   // 8 waves x 2 tiles = 16 tiles
  const int ti0 = t0 >> 2, tj0 = t0 & 3;
  const int ti1 = t1 >> 2, tj1 = t1 & 3;

  // Taylor: for k=2..NTERM: P <- (P*X)/k ; S += P.
  // Results live in registers block-wide after phase 1, so P can be
  // overwritten in place after one barrier (no extra temp buffer).
#pragma unroll 1
  for (int k = 2; k <= NTERM; ++k) {
    __syncthreads();                       // P,X,S writes visible
    v8f c0 = tile_mm64(Ps, Xs, ti0, tj0, lane);
    v8f c1 = tile_mm64(Ps, Xs, ti1, tj1, lane);
    __syncthreads();                       // all reads of Ps complete
    const float f = 1.0f / (float)k;
    tile_store(Ps, ti0, tj0, lane, c0, f);
    tile_store(Ps, ti1, tj1, lane, c1, f);
    tile_acc  (Ss, ti0, tj0, lane, c0, f);
    tile_acc  (Ss, ti1, tj1, lane, c1, f);
  }

  // Squaring: S <- S*S, in place (same barrier trick), NSQUARE times.
#pragma unroll 1
  for (int q = 0; q < NSQUARE; ++q) {
    __syncthreads();
    v8f c0 = tile_mm64(Ss, Ss, ti0, tj0, lane);
    v8f c1 = tile_mm64(Ss, Ss, ti1, tj1, lane);
    __syncthreads();
    tile_store(Ss, ti0, tj0, lane, c0, 1.f);
    tile_store(Ss, ti1, tj1, lane, c1, 1.f);
  }

  __syncthreads();

  // Coalesced writeout of the 64x64 result.
  float* o = out + (size_t)bs * (DH * DH);
#pragma unroll 1
  for (int e = tid; e < DH * DH; e += 256) {
    o[e] = Ss[(e >> 6) * STR + (e & 63)];
  }
}

extern "C" void kernel_launch(void* const* d_in, const int* in_sizes, int n_in,
                              void* d_out, int out_size, void* d_ws, size_t ws_size,
                              hipStream_t stream) {
  const float* r  = (const float*)d_in[0];   // (4, 4096, 2) f32
  const float* p0 = (const float*)d_in[1];   // (2016,) f32
  const float* p1 = (const float*)d_in[2];   // (2016,) f32
  float* out = (float*)d_out;                // (4, 4096, 64, 64) f32

  const int nbs = in_sizes[0] / 2;           // 16384 (b,s) pairs
  structured_rotation_expm_kernel<<<nbs, 256, 0, stream>>>(r, p0, p1, out);
}